// GCNNodeModel_25512105738335
// MI455X (gfx1250) — compile-verified
//
#include <hip/hip_runtime.h>

typedef __attribute__((ext_vector_type(2))) float v2f;
typedef __attribute__((ext_vector_type(8))) float v8f;

#define IN_DIM 128
#define HID 64
#define W1T_STRIDE 132   // 128 + 4 pad: conflict-free ds_load_b64 across the wave

// ---------------------------------------------------------------- zero init
__global__ void zero_f32(float* __restrict__ p, int n) {
  int i = blockIdx.x * blockDim.x + threadIdx.x;
  if (i < n) p[i] = 0.0f;
}

// ---------------------------------------------------------------- layer 1 GEMM
// h1[N,64] = x[N,128] @ W1[128,64] + b1, via V_WMMA_F32_16X16X4_F32.
// One wave per 16x16 output tile; 32 k-steps of K=4, W1 staged transposed in LDS.
// A 16x4 f32 layout: lanes 0-15 -> M=lane,   VGPR0/1 = K0/K1;
//                    lanes16-31 -> M=lane-16, VGPR0/1 = K2/K3 (contiguous float2/lane)
// B 4x16 f32 layout: VGPR0 = K0 (lanes 0-15, N=lane) / K2 (lanes 16-31);  VGPR1 = K1 / K3.
// C/D 16x16 f32:     VGPR v -> M = v + (lane>=16 ? 8 : 0), N = lane&15.
__global__ __launch_bounds__(256) void gemm1_wmma(
    const float* __restrict__ x, const float* __restrict__ W1,
    const float* __restrict__ b1, float* __restrict__ h1, int n_nodes) {
  __shared__ float w1t[HID * W1T_STRIDE];        // 33 KB of the 320 KB WGP LDS

  // cooperative fill: coalesced read of W1[k*64+c], transposed store
  for (int i = threadIdx.x; i < IN_DIM * HID; i += blockDim.x) {
    int k = i >> 6;          // 0..127
    int c = i & 63;          // 0..63
    w1t[c * W1T_STRIDE + k] = W1[i];
  }
  __syncthreads();

  const int lane = threadIdx.x & 31;
  const int wave = blockIdx.x * (blockDim.x >> 5) + (threadIdx.x >> 5);
  const int tiles_n = HID / 16;                  // 4 column tiles
  const int row0 = (wave / tiles_n) * 16;
  const int col0 = (wave % tiles_n) * 16;
  if (row0 >= n_nodes) return;                   // wave-uniform: EXEC all-ones at WMMA

  const int m    = lane & 15;
  const int koff = (lane >> 4) << 1;             // 0 for lanes 0-15, 2 for lanes 16-31
  const int arow = min(row0 + m, n_nodes - 1);
  const float* __restrict__ xrow = x + (size_t)arow * IN_DIM + koff;
  const float* __restrict__ wcol = w1t + (col0 + m) * W1T_STRIDE + koff;

  v8f acc = {};
  #pragma unroll 8
  for (int k0 = 0; k0 < IN_DIM; k0 += 4) {
    v2f a = *(const v2f*)(xrow + k0);            // global_load_b64
    v2f b = *(const v2f*)(wcol + k0);            // ds_load_b64, conflict-free
    acc = __builtin_amdgcn_wmma_f32_16x16x4_f32(
        /*neg_a=*/false, a, /*neg_b=*/false, b,
        /*c_mod=*/(short)0, acc, /*reuse_a=*/false, /*reuse_b=*/false);
  }

  const float bias = b1[col0 + m];
  const int   mrow = row0 + ((lane >> 4) << 3);  // M base for this lane's half
  float* __restrict__ o = h1 + (size_t)mrow * HID + col0 + m;
  if (mrow + 8 <= n_nodes) {                     // wave-uniform fast path (always taken)
    #pragma unroll
    for (int v = 0; v < 8; ++v) o[(size_t)v * HID] = acc[v] + bias;
  } else {
    #pragma unroll
    for (int v = 0; v < 8; ++v)
      if (mrow + v < n_nodes) o[(size_t)v * HID] = acc[v] + bias;
  }
}

// ---------------------------------------------------------------- SpMM (64-wide)
// One wave per edge: gather h[src] (float2 per lane), scale, 2 f32 atomics/lane.
// Edge scalars are wave-uniform -> scalar loads; h is L2-resident (25.6 MB).
__global__ __launch_bounds__(256) void spmm_wide(
    const int* __restrict__ esrc, const int* __restrict__ edst,
    const float* __restrict__ ew, const float* __restrict__ h,
    float* __restrict__ acc, int n_edges) {
  const int e = blockIdx.x * (blockDim.x >> 5) + (threadIdx.x >> 5);
  if (e >= n_edges) return;
  const int   lane = threadIdx.x & 31;
  const int   s    = esrc[e];
  const int   d    = edst[e];
  const float w    = ew[e];
  v2f v = ((const v2f*)(h + (size_t)s * HID))[lane];
  float* out = acc + (size_t)d * HID + lane * 2;
  atomicAdd(out,     w * v.x);   // global_atomic_add_f32, no return
  atomicAdd(out + 1, w * v.y);
}

// ---------------------------------------------------------------- relu + layer 2
// h3[n] = b2 + sum_c relu(h2[n][c]) * W2[c];  one wave per node, shfl_xor tree.
__global__ __launch_bounds__(256) void relu_linear2(
    const float* __restrict__ h2, const float* __restrict__ W2,
    const float* __restrict__ b2, float* __restrict__ h3, int n_nodes) {
  const int n = blockIdx.x * (blockDim.x >> 5) + (threadIdx.x >> 5);
  if (n >= n_nodes) return;
  const int lane = threadIdx.x & 31;
  v2f v  = ((const v2f*)(h2 + (size_t)n * HID))[lane];
  v2f wv = ((const v2f*)W2)[lane];
  float s = fmaxf(v.x, 0.0f) * wv.x + fmaxf(v.y, 0.0f) * wv.y;
  #pragma unroll
  for (int off = 16; off > 0; off >>= 1) s += __shfl_xor(s, off, 32);
  if (lane == 0) h3[n] = s + b2[0];
}

// ---------------------------------------------------------------- final SpMM (scalar)
__global__ __launch_bounds__(256) void spmm_scalar(
    const int* __restrict__ esrc, const int* __restrict__ edst,
    const float* __restrict__ ew, const float* __restrict__ h3,
    float* __restrict__ out, int n_edges) {
  int e = blockIdx.x * blockDim.x + threadIdx.x;
  if (e < n_edges) atomicAdd(&out[edst[e]], ew[e] * h3[esrc[e]]);
}

// ---------------------------------------------------------------- launch
extern "C" void kernel_launch(void* const* d_in, const int* in_sizes, int n_in,
                              void* d_out, int out_size, void* d_ws, size_t ws_size,
                              hipStream_t stream) {
  const float* x    = (const float*)d_in[0];
  const int*   esrc = (const int*)  d_in[1];
  const int*   edst = (const int*)  d_in[2];
  const float* ew   = (const float*)d_in[3];
  const float* W1   = (const float*)d_in[4];
  const float* b1   = (const float*)d_in[5];
  const float* W2   = (const float*)d_in[6];
  const float* b2   = (const float*)d_in[7];
  float*       out  = (float*)d_out;

  const int n_nodes = in_sizes[0] / IN_DIM;   // 100000
  const int n_edges = in_sizes[1];            // 1600000

  float* h1 = (float*)d_ws;                          // [N, 64]
  float* h2 = h1 + (size_t)n_nodes * HID;            // [N, 64] accumulator
  float* h3 = h2 + (size_t)n_nodes * HID;            // [N]

  // zero accumulators (d_out is poisoned by harness; h2 must be 0 every call)
  {
    int n = n_nodes * HID;
    zero_f32<<<(n + 255) / 256, 256, 0, stream>>>(h2, n);
    zero_f32<<<(n_nodes + 255) / 256, 256, 0, stream>>>(out, n_nodes);
  }
  // layer 1: x @ W1 + b1  (WMMA, LDS-staged W1^T)
  {
    const int tiles = ((n_nodes + 15) / 16) * (HID / 16);
    const int wpb   = 8;                      // 8 waves / block (256 threads)
    gemm1_wmma<<<(tiles + wpb - 1) / wpb, 256, 0, stream>>>(x, W1, b1, h1, n_nodes);
  }
  // SpMM 1 (wide)
  spmm_wide<<<(n_edges + 7) / 8, 256, 0, stream>>>(esrc, edst, ew, h1, h2, n_edges);
  // relu + layer 2
  relu_linear2<<<(n_nodes + 7) / 8, 256, 0, stream>>>(h2, W2, b2, h3, n_nodes);
  // SpMM 2 (scalar)
  spmm_scalar<<<(n_edges + 255) / 256, 256, 0, stream>>>(esrc, edst, ew, h3, out, n_edges);
}